// InteractionModule_20418274525614
// MI455X (gfx1250) — compile-verified
//
#include <hip/hip_runtime.h>
#include <hip/hip_bf16.h>
#include <stdint.h>

// ---------------------------------------------------------------------------
// InteractionModule on MI455X (gfx1250, wave32, WMMA f32 16x16x4)
//
//   out[i,j,k] = ( sum_{c,e} p[i,c] * c[j,e] * W3[k,c,e] + b_out[k] ) * mask
//
// Factorized as two f32 WMMA GEMMs:
//   U[(k,c), j]  = Wout_flat[(k,c), e] @ C^T[e, j]     (4096 x 128 x K=32)
//   out[i,(j,k)] = P[i, c]            @ Ut[c, (j,k)]   (640 x 16384 x K=32)
// ---------------------------------------------------------------------------

typedef __attribute__((ext_vector_type(2))) float v2f;
typedef __attribute__((ext_vector_type(8))) float v8f;

#define LP   640
#define LC   128
#define NODE 128
#define HID  32
#define PAIR 128
#define NCOL (LC * PAIR)   // 16384, columns of final GEMM

// ---------------------------------------------------------------------------
// Kernel 1: fused LayerNorm + linear projection for p and c rows.
// One wave (32 lanes) per row; row has 128 elements (4 per lane).
// ---------------------------------------------------------------------------
__global__ void ln_proj_kernel(const float* __restrict__ pe, const float* __restrict__ ce,
                               const float* __restrict__ lnpw, const float* __restrict__ lnpb,
                               const float* __restrict__ lncw, const float* __restrict__ lncb,
                               const float* __restrict__ Wp, const float* __restrict__ bp,
                               const float* __restrict__ Wc, const float* __restrict__ bc,
                               float* __restrict__ Pbuf, float* __restrict__ Cbuf) {
  __shared__ float y[NODE];
  const int r = blockIdx.x;
  const float *x, *lw, *lb, *W, *bb;
  float* o;
  if (r < LP) {
    x = pe + r * NODE; lw = lnpw; lb = lnpb; W = Wp; bb = bp; o = Pbuf + r * HID;
  } else {
    const int rc = r - LP;
    x = ce + rc * NODE; lw = lncw; lb = lncb; W = Wc; bb = bc; o = Cbuf + rc * HID;
  }
  const int lane = threadIdx.x;  // 0..31

  float v0 = x[lane], v1 = x[lane + 32], v2 = x[lane + 64], v3 = x[lane + 96];
  float s = v0 + v1 + v2 + v3;
#pragma unroll
  for (int off = 16; off > 0; off >>= 1) s += __shfl_xor(s, off, 32);
  const float mean = s * (1.0f / 128.0f);

  const float d0 = v0 - mean, d1 = v1 - mean, d2 = v2 - mean, d3 = v3 - mean;
  float sq = d0 * d0 + d1 * d1 + d2 * d2 + d3 * d3;
#pragma unroll
  for (int off = 16; off > 0; off >>= 1) sq += __shfl_xor(sq, off, 32);
  const float rstd = rsqrtf(sq * (1.0f / 128.0f) + 1e-5f);

  y[lane +  0] = d0 * rstd * lw[lane +  0] + lb[lane +  0];
  y[lane + 32] = d1 * rstd * lw[lane + 32] + lb[lane + 32];
  y[lane + 64] = d2 * rstd * lw[lane + 64] + lb[lane + 64];
  y[lane + 96] = d3 * rstd * lw[lane + 96] + lb[lane + 96];
  __syncthreads();

  // lane computes output channel h = lane (HID == 32 == wave width)
  float acc = bb[lane];
  const float* wrow = W + lane * NODE;
#pragma unroll 8
  for (int n = 0; n < NODE; ++n) acc += y[n] * wrow[n];  // y[n] is an LDS broadcast
  o[lane] = acc;
}

// ---------------------------------------------------------------------------
// Kernel 2: mask output (second tuple element): mask[i,j] = p_mask[i] & c_mask[j]
// ---------------------------------------------------------------------------
__global__ void mask_kernel(const unsigned char* __restrict__ pm,
                            const unsigned char* __restrict__ cm,
                            float* __restrict__ mout) {
  const int idx = blockIdx.x * blockDim.x + threadIdx.x;  // 0..81919
  const int i = idx >> 7, j = idx & 127;
  mout[idx] = (pm[i] && cm[j]) ? 1.0f : 0.0f;
}

// ---------------------------------------------------------------------------
// Kernel 3 (GEMM-a): U[(k,c), j] = Wout[(k,c), e] @ C^T[e, j]
//   A = W_out viewed as row-major [4096, 32]   (W_out[k, c*32+e] -> row k*32+c)
//   B[e][j] = Cbuf[j*32 + e]
// Result stored transposed/scattered into Ut[c * 16384 + j*128 + k] so that
// GEMM-b reads it fully coalesced. One 16x16 tile per wave, 8 wmma (K=32).
// ---------------------------------------------------------------------------
__global__ void gemm_a_kernel(const float* __restrict__ Wout,
                              const float* __restrict__ Cbuf,
                              float* __restrict__ Ut) {
  const int lane = threadIdx.x & 31;
  const int wave = threadIdx.x >> 5;
  const int tile = blockIdx.x * 8 + wave;   // 0..2047
  const int tm = tile >> 3;                 // 0..255 (M=4096/16)
  const int tn = tile & 7;                  // 0..7   (N=128/16)
  const int mn   = lane & 15;               // A-row within tile == B-col within tile
  const int half = lane >> 4;               // selects K pair {0,1} vs {2,3}
  const int kq   = half * 2;

  const float* Arow = Wout + (tm * 16 + mn) * 32;   // row-major [4096,32]
  const float* Bcol = Cbuf + (tn * 16 + mn) * 32;   // B[e][j] = C[j,e]

  v8f acc = {};
#pragma unroll
  for (int kb = 0; kb < 32; kb += 4) {
    v2f a, b;
    a.x = Arow[kb + kq];  a.y = Arow[kb + kq + 1];
    b.x = Bcol[kb + kq];  b.y = Bcol[kb + kq + 1];
    acc = __builtin_amdgcn_wmma_f32_16x16x4_f32(false, a, false, b, (short)0, acc,
                                                false, false);
  }

  const int j = tn * 16 + mn;
#pragma unroll
  for (int v = 0; v < 8; ++v) {
    const int row = tm * 16 + v + half * 8;  // row = k*32 + c
    const int k = row >> 5, c = row & 31;
    Ut[c * NCOL + j * PAIR + k] = acc[v];    // scatter once; 2MB stays in L2
  }
}

// ---------------------------------------------------------------------------
// Kernel 4 (GEMM-b): out[i, n] = P[i, c] @ Ut[c, n] + b_out[n&127], masked.
//   n = j*128 + k  ==> output is already in reference flat layout [i, j, k].
// One 16x16 tile per wave, 8 wmma (K=32). A from 80KB P (L2), B coalesced
// from 2MB Ut (L2). 42MB output written with non-temporal stores.
// ---------------------------------------------------------------------------
__global__ void gemm_b_kernel(const float* __restrict__ Pbuf,
                              const float* __restrict__ Ut,
                              const float* __restrict__ b_out,
                              const unsigned char* __restrict__ pm,
                              const unsigned char* __restrict__ cm,
                              float* __restrict__ out) {
  const int lane = threadIdx.x & 31;
  const int wave = threadIdx.x >> 5;
  const int tile = blockIdx.x * 8 + wave;   // 0..40959
  const int tm = tile >> 10;                // 0..39   (M=640/16)
  const int tn = tile & 1023;               // 0..1023 (N=16384/16)
  const int mn   = lane & 15;
  const int half = lane >> 4;
  const int kq   = half * 2;

  const float* Arow = Pbuf + (tm * 16 + mn) * HID;
  const int n0 = tn * 16;

  v8f acc = {};
#pragma unroll
  for (int kb = 0; kb < 32; kb += 4) {
    v2f a, b;
    a.x = Arow[kb + kq];  a.y = Arow[kb + kq + 1];
    const float* B0 = Ut + (kb + kq) * NCOL + n0 + mn;  // coalesced 64B per half-wave
    b.x = B0[0];          b.y = B0[NCOL];
    acc = __builtin_amdgcn_wmma_f32_16x16x4_f32(false, a, false, b, (short)0, acc,
                                                false, false);
  }

  const int n = n0 + mn;
  const int j = n >> 7, k = n & 127;
  const float bias = b_out[k];
  const float cmf = cm[j] ? 1.0f : 0.0f;
#pragma unroll
  for (int v = 0; v < 8; ++v) {
    const int i = tm * 16 + v + half * 8;
    const float m = pm[i] ? cmf : 0.0f;
    // write-once 42MB stream: bypass cache with non-temporal store
    __builtin_nontemporal_store((acc[v] + bias) * m, &out[i * NCOL + n]);
  }
}

// ---------------------------------------------------------------------------
// Host-side launch
// ---------------------------------------------------------------------------
extern "C" void kernel_launch(void* const* d_in, const int* in_sizes, int n_in,
                              void* d_out, int out_size, void* d_ws, size_t ws_size,
                              hipStream_t stream) {
  const float* p_embed = (const float*)d_in[0];
  const float* c_embed = (const float*)d_in[1];
  const unsigned char* p_mask = (const unsigned char*)d_in[2];
  const unsigned char* c_mask = (const unsigned char*)d_in[3];
  const float* ln_p_w = (const float*)d_in[4];
  const float* ln_p_b = (const float*)d_in[5];
  const float* ln_c_w = (const float*)d_in[6];
  const float* ln_c_b = (const float*)d_in[7];
  const float* W_p   = (const float*)d_in[8];
  const float* b_p   = (const float*)d_in[9];
  const float* W_c   = (const float*)d_in[10];
  const float* b_c   = (const float*)d_in[11];
  const float* W_out = (const float*)d_in[12];
  const float* b_out = (const float*)d_in[13];

  float* out  = (float*)d_out;                     // [640, 16384]
  float* mout = out + (size_t)LP * NCOL;           // [640, 128] mask (tuple elem 2)

  // Workspace layout (~2.2 MB total)
  float* Pbuf = (float*)d_ws;                      // [640, 32]
  float* Cbuf = Pbuf + LP * HID;                   // [128, 32]
  float* Utb  = Cbuf + LC * HID;                   // [32, 16384]

  // 1) LayerNorm + projection for all 640 p-rows and 128 c-rows (1 wave/row)
  ln_proj_kernel<<<LP + LC, 32, 0, stream>>>(p_embed, c_embed,
                                             ln_p_w, ln_p_b, ln_c_w, ln_c_b,
                                             W_p, b_p, W_c, b_c, Pbuf, Cbuf);

  // 2) inter_mask output
  mask_kernel<<<(LP * LC) / 256, 256, 0, stream>>>(p_mask, c_mask, mout);

  // 3) GEMM-a: Ut[c, (j,k)] <- W_out[(k,c), e] @ C^T   (2048 tiles, 8 waves/blk)
  gemm_a_kernel<<<256, 256, 0, stream>>>(W_out, Cbuf, Utb);

  // 4) GEMM-b: out[i, (j,k)] <- P @ Ut + bias, masked  (40960 tiles)
  gemm_b_kernel<<<5120, 256, 0, stream>>>(Pbuf, Utb, b_out, p_mask, c_mask, out);
}